// GCNMod_46162308497999
// MI455X (gfx1250) — compile-verified
//
#include <hip/hip_runtime.h>
#include <math.h>

// GCN forward on MI455X (gfx1250):
//   h  = relu(A @ (x @ W1) + b1)
//   out = log_softmax(A @ (h @ W2) + b2)
// Dense GEMMs: f32 WMMA 16x16x4 (exact precision; GEMMs are not the bottleneck).
// Sparse aggregation: dst-CSR built per call (int atomics only), then pure
// gather+FMA aggregation — no float atomics. All intermediates are L2-resident
// (192 MB L2), so aggregation runs at L2 streaming rates instead of atomic-RMW
// rates.

typedef float v2f __attribute__((ext_vector_type(2)));
typedef float v8f __attribute__((ext_vector_type(8)));

#define NFEAT 256
#define NHID  128
#define NCLASS 16

// ---------------------------------------------------------------------------
// GEMM1: h0[N,128] = x[N,256] @ W1[256,128]
// One wave computes a 16x128 strip (8 WMMA 16x16 accumulators).
// f32 WMMA fragment layouts (ISA 7.12.2):
//   A 16x4 : lane = M (mod 16); VGPR0/1 = K = 2*half, 2*half+1 (half = lane>>4)
//   B 4x16 : lane = N (mod 16); VGPR0/1 = K = 2*half, 2*half+1
//   C 16x16: VGPR r: lanes 0-15 -> (M=r, N=lane), lanes 16-31 -> (M=r+8)
// ---------------------------------------------------------------------------
__global__ void gcn_gemm1_wmma(const float* __restrict__ x,
                               const float* __restrict__ W1,
                               float* __restrict__ h0, int n) {
  const int wave   = (blockIdx.x * blockDim.x + threadIdx.x) >> 5;
  const int lane   = threadIdx.x & 31;
  const int mtiles = n >> 4;                 // n % 16 == 0 (N = 100000)
  if (wave >= mtiles) return;                // uniform per wave
  const int half = lane >> 4;
  const int l16  = lane & 15;
  const int row  = wave * 16 + l16;
  const float* __restrict__ xrow = x + (size_t)row * NFEAT;

  v8f acc[8];
#pragma unroll
  for (int t = 0; t < 8; ++t) acc[t] = (v8f){0.f,0.f,0.f,0.f,0.f,0.f,0.f,0.f};

  for (int k = 0; k < NFEAT; k += 4) {
    v2f a;
    a.x = xrow[k + 2 * half + 0];
    a.y = xrow[k + 2 * half + 1];
#pragma unroll
    for (int t = 0; t < 8; ++t) {
      const float* __restrict__ wp =
          W1 + (size_t)(k + 2 * half) * NHID + t * 16 + l16;
      v2f b;
      b.x = wp[0];
      b.y = wp[NHID];
      acc[t] = __builtin_amdgcn_wmma_f32_16x16x4_f32(
          false, a, false, b, (short)0, acc[t], false, false);
    }
  }

#pragma unroll
  for (int t = 0; t < 8; ++t) {
#pragma unroll
    for (int r = 0; r < 8; ++r) {
      const int orow = wave * 16 + r + half * 8;
      h0[(size_t)orow * NHID + t * 16 + l16] = acc[t][r];
    }
  }
}

// ---------------------------------------------------------------------------
// CSR build, step 1: degree histogram over dst (integer atomics only)
// ---------------------------------------------------------------------------
__global__ void gcn_degree(const int* __restrict__ dst, int* __restrict__ deg,
                           int nedges) {
  const int e = blockIdx.x * blockDim.x + threadIdx.x;
  if (e >= nedges) return;
  atomicAdd(&deg[dst[e]], 1);
}

// ---------------------------------------------------------------------------
// CSR build, step 2: exclusive scan of deg -> rowstart[n+1].
// Also rewrites deg[i] with the row start so it can serve as the scatter
// cursor (saves a buffer). Single 1024-thread block, tiled Hillis-Steele.
// ---------------------------------------------------------------------------
__global__ void gcn_scan(int* __restrict__ deg, int* __restrict__ rowstart,
                         int n) {
  __shared__ int smem[1024];
  __shared__ int running;
  if (threadIdx.x == 0) running = 0;
  __syncthreads();
  for (int base = 0; base < n; base += 1024) {
    const int i = base + (int)threadIdx.x;
    const int v = (i < n) ? deg[i] : 0;
    smem[threadIdx.x] = v;
    __syncthreads();
#pragma unroll
    for (int off = 1; off < 1024; off <<= 1) {
      const int t = (threadIdx.x >= (unsigned)off) ? smem[threadIdx.x - off] : 0;
      __syncthreads();
      smem[threadIdx.x] += t;
      __syncthreads();
    }
    const int excl = smem[threadIdx.x] - v;
    if (i < n) {
      const int start = running + excl;
      rowstart[i] = start;
      deg[i]      = start;   // becomes the scatter cursor
    }
    __syncthreads();
    if (threadIdx.x == 1023) running += smem[1023];
    __syncthreads();
  }
  if (threadIdx.x == 0) rowstart[n] = running;
}

// ---------------------------------------------------------------------------
// CSR build, step 3: scatter edges into dst-sorted order
// ---------------------------------------------------------------------------
__global__ void gcn_scatter(const int* __restrict__ src,
                            const int* __restrict__ dst,
                            const float* __restrict__ ew,
                            int* __restrict__ cursor,
                            int* __restrict__ csr_src,
                            float* __restrict__ csr_w, int nedges) {
  const int e = blockIdx.x * blockDim.x + threadIdx.x;
  if (e >= nedges) return;
  const int d   = dst[e];
  const int pos = atomicAdd(&cursor[d], 1);
  csr_src[pos] = src[e];
  csr_w[pos]   = ew[e];
}

// ---------------------------------------------------------------------------
// Layer-1 aggregation (gather form): h[d] = relu(sum_j w_j * h0[src_j] + b1)
// One wave per dst node; each lane owns a float4 (128 features / 32 lanes).
// 512B coalesced gather per edge, register accumulation, one store per row.
// ---------------------------------------------------------------------------
__global__ void gcn_agg1(const float* __restrict__ h0,
                         const int* __restrict__ rowstart,
                         const int* __restrict__ csr_src,
                         const float* __restrict__ csr_w,
                         const float* __restrict__ b1,
                         float* __restrict__ h, int n) {
  const int node = (blockIdx.x * blockDim.x + threadIdx.x) >> 5;
  const int lane = threadIdx.x & 31;
  if (node >= n) return;
  const int beg = rowstart[node];
  const int end = rowstart[node + 1];
  float4 acc = {0.f, 0.f, 0.f, 0.f};
  for (int j = beg; j < end; ++j) {
    const int   s = csr_src[j];
    const float w = csr_w[j];
    const float4 v = ((const float4*)(h0 + (size_t)s * NHID))[lane];
    acc.x += w * v.x;
    acc.y += w * v.y;
    acc.z += w * v.z;
    acc.w += w * v.w;
  }
  const float4 bb = ((const float4*)b1)[lane];
  float4 r;
  r.x = fmaxf(acc.x + bb.x, 0.f);
  r.y = fmaxf(acc.y + bb.y, 0.f);
  r.z = fmaxf(acc.z + bb.z, 0.f);
  r.w = fmaxf(acc.w + bb.w, 0.f);
  ((float4*)(h + (size_t)node * NHID))[lane] = r;
}

// ---------------------------------------------------------------------------
// GEMM2: h2[N,16] = h[N,128] @ W2[128,16]
// One wave per 16x16 output tile, K = 128, f32 WMMA.
// ---------------------------------------------------------------------------
__global__ void gcn_gemm2_wmma(const float* __restrict__ h,
                               const float* __restrict__ W2,
                               float* __restrict__ h2, int n) {
  const int wave   = (blockIdx.x * blockDim.x + threadIdx.x) >> 5;
  const int lane   = threadIdx.x & 31;
  const int mtiles = n >> 4;
  if (wave >= mtiles) return;
  const int half = lane >> 4;
  const int l16  = lane & 15;
  const int row  = wave * 16 + l16;
  const float* __restrict__ arow = h + (size_t)row * NHID;

  v8f acc = (v8f){0.f,0.f,0.f,0.f,0.f,0.f,0.f,0.f};

  for (int k = 0; k < NHID; k += 4) {
    const int k0 = k + 2 * half;
    v2f a;
    a.x = arow[k0 + 0];
    a.y = arow[k0 + 1];
    v2f b;
    b.x = W2[(size_t)(k0 + 0) * NCLASS + l16];
    b.y = W2[(size_t)(k0 + 1) * NCLASS + l16];
    acc = __builtin_amdgcn_wmma_f32_16x16x4_f32(
        false, a, false, b, (short)0, acc, false, false);
  }

#pragma unroll
  for (int r = 0; r < 8; ++r) {
    const int orow = wave * 16 + r + half * 8;
    h2[(size_t)orow * NCLASS + l16] = acc[r];
  }
}

// ---------------------------------------------------------------------------
// Layer-2 aggregation (gather form): out[d][f] = sum_j w_j * h2[src_j][f]
// One lane per (node, class); 16 classes -> two nodes per wave.
// ---------------------------------------------------------------------------
__global__ void gcn_agg2(const float* __restrict__ h2,
                         const int* __restrict__ rowstart,
                         const int* __restrict__ csr_src,
                         const float* __restrict__ csr_w,
                         float* __restrict__ out, int n) {
  const int t    = blockIdx.x * blockDim.x + threadIdx.x;
  const int node = t >> 4;
  const int f    = t & 15;
  if (node >= n) return;
  const int beg = rowstart[node];
  const int end = rowstart[node + 1];
  float acc = 0.f;
  for (int j = beg; j < end; ++j)
    acc += csr_w[j] * h2[(size_t)csr_src[j] * NCLASS + f];
  out[(size_t)node * NCLASS + f] = acc;
}

// ---------------------------------------------------------------------------
// Finalize: logits = out + b2; out = logits - logsumexp(logits)  (in place)
// ---------------------------------------------------------------------------
__global__ void gcn_logsoftmax(float* __restrict__ out,
                               const float* __restrict__ b2, int n) {
  const int r = blockIdx.x * blockDim.x + threadIdx.x;
  if (r >= n) return;
  float* p = out + (size_t)r * NCLASS;
  float v[NCLASS];
  float mx = -INFINITY;
#pragma unroll
  for (int i = 0; i < NCLASS; ++i) {
    v[i] = p[i] + b2[i];
    mx = fmaxf(mx, v[i]);
  }
  float s = 0.f;
#pragma unroll
  for (int i = 0; i < NCLASS; ++i) s += __expf(v[i] - mx);
  const float lse = mx + __logf(s);
#pragma unroll
  for (int i = 0; i < NCLASS; ++i) p[i] = v[i] - lse;
}

extern "C" void kernel_launch(void* const* d_in, const int* in_sizes, int n_in,
                              void* d_out, int out_size, void* d_ws, size_t ws_size,
                              hipStream_t stream) {
  const float* x   = (const float*)d_in[0];
  const float* W1  = (const float*)d_in[1];
  const float* b1  = (const float*)d_in[2];
  const float* W2  = (const float*)d_in[3];
  const float* b2  = (const float*)d_in[4];
  const float* ew  = (const float*)d_in[5];
  const int*   src = (const int*)d_in[6];
  const int*   dst = (const int*)d_in[7];
  float* out = (float*)d_out;

  const int n = in_sizes[0] / NFEAT;   // 100000
  const int E = in_sizes[5];           // 3200000

  // workspace layout (floats/ints, 4B each):
  //   h0 [n*128] | h [n*128] | h2 [n*16] | deg/cursor [n] | rowstart [n+1]
  //   | csr_src [E] | csr_w [E]
  float* h0 = (float*)d_ws;
  float* h  = h0 + (size_t)n * NHID;
  float* h2 = h  + (size_t)n * NHID;
  int*   deg      = (int*)(h2 + (size_t)n * NCLASS);
  int*   rowstart = deg + n;
  int*   csr_src  = rowstart + (n + 1);
  float* csr_w    = (float*)(csr_src + E);

  hipMemsetAsync(deg, 0, (size_t)n * sizeof(int), stream);

  const int mtiles = (n + 15) / 16;

  // Dense x@W1 (independent of CSR build; stream-serialized anyway)
  gcn_gemm1_wmma<<<dim3((mtiles * 32 + 255) / 256), dim3(256), 0, stream>>>(
      x, W1, h0, n);

  // Build dst-CSR (int atomics only)
  gcn_degree<<<dim3((E + 255) / 256), dim3(256), 0, stream>>>(dst, deg, E);
  gcn_scan<<<dim3(1), dim3(1024), 0, stream>>>(deg, rowstart, n);
  gcn_scatter<<<dim3((E + 255) / 256), dim3(256), 0, stream>>>(
      src, dst, ew, deg, csr_src, csr_w, E);

  // Layer 1 aggregation + bias + relu (gather, no float atomics)
  gcn_agg1<<<dim3((n * 32 + 255) / 256), dim3(256), 0, stream>>>(
      h0, rowstart, csr_src, csr_w, b1, h, n);

  // Dense h@W2
  gcn_gemm2_wmma<<<dim3((mtiles * 32 + 255) / 256), dim3(256), 0, stream>>>(
      h, W2, h2, n);

  // Layer 2 aggregation (gather)
  gcn_agg2<<<dim3((n * NCLASS + 255) / 256), dim3(256), 0, stream>>>(
      h2, rowstart, csr_src, csr_w, out, n);

  // bias + log_softmax, in place
  gcn_logsoftmax<<<dim3((n + 255) / 256), dim3(256), 0, stream>>>(out, b2, n);
}